// GASConv_90305982366105
// MI455X (gfx1250) — compile-verified
//
#include <hip/hip_runtime.h>
#include <hip/hip_bf16.h>
#include <math.h>

typedef __attribute__((ext_vector_type(2))) float v2f;
typedef __attribute__((ext_vector_type(8))) float v8f;

// ---------------------------------------------------------------------------
// Float atomic max via int/uint ordering trick (works for mixed signs).
// ---------------------------------------------------------------------------
__device__ __forceinline__ void atomic_max_f32(float* addr, float val) {
    if (val >= 0.0f) {
        atomicMax((int*)addr, __float_as_int(val));
    } else {
        atomicMin((unsigned int*)addr, __float_as_uint(val));
    }
}

// ---------------------------------------------------------------------------
// Generic WMMA f32 GEMM:  Y[:, col_off : col_off+NT*16] = act(X @ W + bias)
//   X: [M,K] row-major, W: [K, NT*16] row-major, Y has leading dim ldy.
// One wave per 16-row tile. K stepped by 4 using v_wmma_f32_16x16x4_f32.
// A layout per lane: m = lane&15, kk = lane>>4 ; a = {X[m][k+2kk], X[m][k+2kk+1]}
// B layout per lane: n = lane&15, kk = lane>>4 ; b = {W[k+2kk][n], W[k+2kk+1][n]}
// D layout: vgpr j holds row (j + 8*kk), col = lane&15.
// ---------------------------------------------------------------------------
template <int NT>
__global__ void gemm_bias_kernel(const float* __restrict__ X,
                                 const float* __restrict__ W,
                                 const float* __restrict__ bias,
                                 float* __restrict__ Y,
                                 int M, int K, int ldy, int col_off, int do_relu) {
    const int wave = threadIdx.x >> 5;
    const int lane = threadIdx.x & 31;
    const int tile = blockIdx.x * (blockDim.x >> 5) + wave;
    if (tile * 16 >= M) return;                 // uniform across wave

    const int m  = lane & 15;
    const int kk = lane >> 4;
    int row = tile * 16 + m;
    if (row >= M) row = M - 1;                  // uniform-safe clamp (read only)
    const float* xrow = X + (size_t)row * K;
    const int Nc = NT * 16;

    v8f acc[NT] = {};
    for (int k = 0; k < K; k += 4) {
        v2f a;
        a.x = xrow[k + 2 * kk];
        a.y = xrow[k + 2 * kk + 1];
        const float* wr0 = W + (size_t)(k + 2 * kk) * Nc;
        const float* wr1 = wr0 + Nc;
#pragma unroll
        for (int nt = 0; nt < NT; ++nt) {
            v2f b;
            b.x = wr0[nt * 16 + m];
            b.y = wr1[nt * 16 + m];
            acc[nt] = __builtin_amdgcn_wmma_f32_16x16x4_f32(
                false, a, false, b, (short)0, acc[nt], false, false);
        }
    }

    float bv[NT];
#pragma unroll
    for (int nt = 0; nt < NT; ++nt) bv[nt] = bias[nt * 16 + m];

    if (tile * 16 + 16 <= M) {
        // Full tile: no per-element bounds check (uniform fast path).
#pragma unroll
        for (int j = 0; j < 8; ++j) {
            const int r = tile * 16 + j + 8 * kk;
            float* yrow = Y + (size_t)r * ldy + col_off + m;
#pragma unroll
            for (int nt = 0; nt < NT; ++nt) {
                float v = acc[nt][j] + bv[nt];
                if (do_relu) v = fmaxf(v, 0.0f);
                yrow[nt * 16] = v;
            }
        }
    } else {
#pragma unroll
        for (int j = 0; j < 8; ++j) {
            const int r = tile * 16 + j + 8 * kk;
            if (r < M) {
                float* yrow = Y + (size_t)r * ldy + col_off + m;
#pragma unroll
                for (int nt = 0; nt < NT; ++nt) {
                    float v = acc[nt][j] + bv[nt];
                    if (do_relu) v = fmaxf(v, 0.0f);
                    yrow[nt * 16] = v;
                }
            }
        }
    }
}

// ---------------------------------------------------------------------------
// Fused edge kernel:
//   Out[e,:] = relu( Xe[e,:] @ W + bias + TabA[idxA[e],:] + TabB[idxB[e],:] )
// Xe: [E,64] edge features, W: w_e [64,64], TabA/TabB: [N,64].
// ---------------------------------------------------------------------------
__global__ void edge_gemm_fused_kernel(const float* __restrict__ Xe,
                                       const float* __restrict__ W,
                                       const float* __restrict__ bias,
                                       const float* __restrict__ TabA,
                                       const float* __restrict__ TabB,
                                       const int* __restrict__ idxA,
                                       const int* __restrict__ idxB,
                                       float* __restrict__ Out,
                                       int E) {
    const int wave = threadIdx.x >> 5;
    const int lane = threadIdx.x & 31;
    const int tile = blockIdx.x * (blockDim.x >> 5) + wave;
    if (tile * 16 >= E) return;

    const int m  = lane & 15;
    const int kk = lane >> 4;
    int row = tile * 16 + m;
    if (row >= E) row = E - 1;
    const float* xrow = Xe + (size_t)row * 64;

    v8f acc[4] = {};
    for (int k = 0; k < 64; k += 4) {
        v2f a;
        a.x = xrow[k + 2 * kk];
        a.y = xrow[k + 2 * kk + 1];
        const float* wr0 = W + (size_t)(k + 2 * kk) * 64;
        const float* wr1 = wr0 + 64;
#pragma unroll
        for (int nt = 0; nt < 4; ++nt) {
            v2f b;
            b.x = wr0[nt * 16 + m];
            b.y = wr1[nt * 16 + m];
            acc[nt] = __builtin_amdgcn_wmma_f32_16x16x4_f32(
                false, a, false, b, (short)0, acc[nt], false, false);
        }
    }

    float bv[4];
#pragma unroll
    for (int nt = 0; nt < 4; ++nt) bv[nt] = bias[nt * 16 + m];

    if (tile * 16 + 16 <= E) {
        // Full tile fast path: indices loaded once per row, unguarded stores.
#pragma unroll
        for (int j = 0; j < 8; ++j) {
            const int r = tile * 16 + j + 8 * kk;
            const int ia = idxA[r];
            const int ib = idxB[r];
            const float* ta = TabA + (size_t)ia * 64 + m;
            const float* tb = TabB + (size_t)ib * 64 + m;
            float* orow = Out + (size_t)r * 64 + m;
#pragma unroll
            for (int nt = 0; nt < 4; ++nt) {
                float v = acc[nt][j] + bv[nt] + ta[nt * 16] + tb[nt * 16];
                orow[nt * 16] = fmaxf(v, 0.0f);
            }
        }
    } else {
#pragma unroll
        for (int j = 0; j < 8; ++j) {
            const int r = tile * 16 + j + 8 * kk;
            if (r < E) {
                const int ia = idxA[r];
                const int ib = idxB[r];
                const float* ta = TabA + (size_t)ia * 64 + m;
                const float* tb = TabB + (size_t)ib * 64 + m;
                float* orow = Out + (size_t)r * 64 + m;
#pragma unroll
                for (int nt = 0; nt < 4; ++nt) {
                    float v = acc[nt][j] + bv[nt] + ta[nt * 16] + tb[nt * 16];
                    orow[nt * 16] = fmaxf(v, 0.0f);
                }
            }
        }
    }
}

// ---------------------------------------------------------------------------
// Segment-softmax attention passes. One thread per (edge, channel), c in [0,128).
//   x   = h_ve[c] * h_att[src][c],  h_ve[c] = c<64 ? node[src][c] : edge[e][c-64]
// MODE 0: mbuf[dst][c] = max(x)         (atomic float max)
// MODE 1: sbuf[dst][c] += exp(x - m)
// MODE 2: agg [dst][c] += h_ve[c] * exp(x - m) / s
// ---------------------------------------------------------------------------
template <int MODE>
__global__ void att_pass_kernel(const float* __restrict__ node_feat,
                                const float* __restrict__ edge_feat,
                                const float* __restrict__ h_att,
                                const int* __restrict__ src,
                                const int* __restrict__ dst,
                                float* __restrict__ mbuf,
                                float* __restrict__ sbuf,
                                float* __restrict__ agg,
                                int E) {
    const long long tid   = (long long)blockIdx.x * blockDim.x + threadIdx.x;
    const long long total = (long long)E * 128;
    if (tid >= total) return;
    const int e = (int)(tid >> 7);
    const int c = (int)(tid & 127);
    const int s = src[e];
    const int d = dst[e];

    const float hv = (c < 64) ? node_feat[(size_t)s * 64 + c]
                              : edge_feat[(size_t)e * 64 + (c - 64)];
    const float x  = hv * h_att[(size_t)s * 128 + c];
    const size_t o = (size_t)d * 128 + c;

    if (MODE == 0) {
        atomic_max_f32(&mbuf[o], x);
    } else if (MODE == 1) {
        atomicAdd(&sbuf[o], expf(x - mbuf[o]));
    } else {
        const float w = expf(x - mbuf[o]) / sbuf[o];
        atomicAdd(&agg[o], hv * w);
    }
}

__global__ void fill_kernel(float* __restrict__ p, float val, size_t n) {
    size_t i = (size_t)blockIdx.x * blockDim.x + threadIdx.x;
    if (i < n) p[i] = val;
}

// ---------------------------------------------------------------------------
extern "C" void kernel_launch(void* const* d_in, const int* in_sizes, int n_in,
                              void* d_out, int out_size, void* d_ws, size_t ws_size,
                              hipStream_t stream) {
    const float* f_feat = (const float*)d_in[0];
    const float* b_feat = (const float*)d_in[1];
    const float* u_feat = (const float*)d_in[2];
    const float* v_feat = (const float*)d_in[3];
    const int*   fw_src = (const int*)d_in[4];
    const int*   fw_dst = (const int*)d_in[5];
    const int*   bw_src = (const int*)d_in[6];
    const int*   bw_dst = (const int*)d_in[7];
    const float* W_e  = (const float*)d_in[8];
    const float* B_e  = (const float*)d_in[9];
    const float* W_u  = (const float*)d_in[10];
    const float* B_u  = (const float*)d_in[11];
    const float* W_v  = (const float*)d_in[12];
    const float* B_v  = (const float*)d_in[13];
    const float* W_au = (const float*)d_in[14];
    const float* B_au = (const float*)d_in[15];
    const float* W_av = (const float*)d_in[16];
    const float* B_av = (const float*)d_in[17];
    const float* W_Wu = (const float*)d_in[18];
    const float* B_Wu = (const float*)d_in[19];
    const float* W_Wv = (const float*)d_in[20];
    const float* B_Wv = (const float*)d_in[21];
    const float* W_Vu = (const float*)d_in[22];
    const float* B_Vu = (const float*)d_in[23];
    const float* W_Vv = (const float*)d_in[24];
    const float* B_Vv = (const float*)d_in[25];

    const int E = in_sizes[0] / 64;
    const int N = in_sizes[2] / 64;

    float* out = (float*)d_out;
    float* hf = out;                                   // [E,64]
    float* hb = out + (size_t)E * 64;                  // [E,64]
    float* hu = out + (size_t)2 * E * 64;              // [N,128]
    float* hv = hu + (size_t)N * 128;                  // [N,128]

    // Workspace carve-up (floats)
    float* ws = (float*)d_ws;
    size_t o = 0;
    float* src_he_u = ws + o; o += (size_t)N * 64;
    float* src_he_v = ws + o; o += (size_t)N * 64;
    float* dst_he_v = ws + o; o += (size_t)N * 64;
    float* h_att_u  = ws + o; o += (size_t)N * 128;
    float* h_att_v  = ws + o; o += (size_t)N * 128;
    float* m_bw     = ws + o; o += (size_t)N * 128;
    float* m_fw     = ws + o; o += (size_t)N * 128;
    float* s_bw     = ws + o; o += (size_t)N * 128;
    float* s_fw     = ws + o; o += (size_t)N * 128;
    float* agg_u    = ws + o; o += (size_t)N * 128;
    float* agg_v    = ws + o; o += (size_t)N * 128;

    const int TPB = 256;               // 8 waves / block
    const int TILES_PB = TPB / 32;
    const int nodeTiles = (N + 15) / 16;
    const int edgeTiles = (E + 15) / 16;
    const int nodeGrid = (nodeTiles + TILES_PB - 1) / TILES_PB;
    const int edgeGrid = (edgeTiles + TILES_PB - 1) / TILES_PB;

    // 1) Node projections (WMMA)
    gemm_bias_kernel<4><<<nodeGrid, TPB, 0, stream>>>(u_feat, W_u,  B_u,  src_he_u, N, 64,  64,  0, 0);
    gemm_bias_kernel<4><<<nodeGrid, TPB, 0, stream>>>(v_feat, W_v,  B_v,  src_he_v, N, 64,  64,  0, 0);
    gemm_bias_kernel<4><<<nodeGrid, TPB, 0, stream>>>(v_feat, W_u,  B_u,  dst_he_v, N, 64,  64,  0, 0);
    gemm_bias_kernel<8><<<nodeGrid, TPB, 0, stream>>>(u_feat, W_au, B_au, h_att_u,  N, 64,  128, 0, 0);
    gemm_bias_kernel<8><<<nodeGrid, TPB, 0, stream>>>(v_feat, W_av, B_av, h_att_v,  N, 64,  128, 0, 0);

    // 2) Fused edge GEMM + gather-add-relu (WMMA)
    // hf = relu(b_feat@W_e + B_e + src_he_u[bw_src] + src_he_v[bw_dst])
    edge_gemm_fused_kernel<<<edgeGrid, TPB, 0, stream>>>(
        b_feat, W_e, B_e, src_he_u, src_he_v, bw_src, bw_dst, hf, E);
    // hb = relu(f_feat@W_e + B_e + src_he_u[fw_src] + dst_he_v[fw_dst])
    edge_gemm_fused_kernel<<<edgeGrid, TPB, 0, stream>>>(
        f_feat, W_e, B_e, src_he_u, dst_he_v, fw_src, fw_dst, hb, E);

    // 3) Init softmax buffers: m -> -inf, s/agg -> 0
    {
        size_t nM = (size_t)N * 128 * 2;   // m_bw, m_fw (contiguous)
        size_t nZ = (size_t)N * 128 * 4;   // s_bw, s_fw, agg_u, agg_v (contiguous)
        fill_kernel<<<(unsigned)((nM + TPB - 1) / TPB), TPB, 0, stream>>>(m_bw, -INFINITY, nM);
        fill_kernel<<<(unsigned)((nZ + TPB - 1) / TPB), TPB, 0, stream>>>(s_bw, 0.0f, nZ);
    }

    // 4) Segment softmax + weighted aggregation (3 passes each direction)
    {
        long long total = (long long)E * 128;
        unsigned grid = (unsigned)((total + TPB - 1) / TPB);
        // backward direction -> agg_u
        att_pass_kernel<0><<<grid, TPB, 0, stream>>>(v_feat, b_feat, h_att_u, bw_src, bw_dst,
                                                     m_bw, s_bw, agg_u, E);
        att_pass_kernel<1><<<grid, TPB, 0, stream>>>(v_feat, b_feat, h_att_u, bw_src, bw_dst,
                                                     m_bw, s_bw, agg_u, E);
        att_pass_kernel<2><<<grid, TPB, 0, stream>>>(v_feat, b_feat, h_att_u, bw_src, bw_dst,
                                                     m_bw, s_bw, agg_u, E);
        // forward direction -> agg_v
        att_pass_kernel<0><<<grid, TPB, 0, stream>>>(u_feat, f_feat, h_att_v, fw_src, fw_dst,
                                                     m_fw, s_fw, agg_v, E);
        att_pass_kernel<1><<<grid, TPB, 0, stream>>>(u_feat, f_feat, h_att_v, fw_src, fw_dst,
                                                     m_fw, s_fw, agg_v, E);
        att_pass_kernel<2><<<grid, TPB, 0, stream>>>(u_feat, f_feat, h_att_v, fw_src, fw_dst,
                                                     m_fw, s_fw, agg_v, E);
    }

    // 5) Output node features (concat done via ldy=128 + column offset)
    gemm_bias_kernel<4><<<nodeGrid, TPB, 0, stream>>>(u_feat, W_Vu, B_Vu, hu, N, 64,  128, 0,  0);
    gemm_bias_kernel<4><<<nodeGrid, TPB, 0, stream>>>(agg_u,  W_Wu, B_Wu, hu, N, 128, 128, 64, 1);
    gemm_bias_kernel<4><<<nodeGrid, TPB, 0, stream>>>(v_feat, W_Vv, B_Vv, hv, N, 64,  128, 0,  0);
    gemm_bias_kernel<4><<<nodeGrid, TPB, 0, stream>>>(agg_v,  W_Wv, B_Wv, hv, N, 128, 128, 64, 1);
}